// PatchoutMask_67559835566231
// MI455X (gfx1250) — compile-verified
//
#include <hip/hip_runtime.h>
#include <stdint.h>

// Problem constants (reference: B,T,D = 16, 2048, 768)
#define PB 16
#define PT 2048
#define PD 768

// ---------------------------------------------------------------------------
// CDNA5 async global->LDS helpers (gfx1250). Inline asm path is portable
// across ROCm 7.2 / amdgpu-toolchain per the bridge doc; the wait uses the
// builtin when the compiler declares it.
// ---------------------------------------------------------------------------
__device__ __forceinline__ void async_load_b128_to_lds(uint32_t lds_byte_addr,
                                                       const void* gaddr) {
    // VDST = per-lane LDS byte address, VADDR = 64-bit global address (GV mode)
    asm volatile("global_load_async_to_lds_b128 %0, %1, off"
                 :
                 : "v"(lds_byte_addr), "v"(gaddr)
                 : "memory");
}

__device__ __forceinline__ void wait_async_cnt0() {
#if __has_builtin(__builtin_amdgcn_s_wait_asynccnt)
    __builtin_amdgcn_s_wait_asynccnt(0);
#else
    asm volatile("s_wait_asynccnt 0x0" ::: "memory");
#endif
}

// ---------------------------------------------------------------------------
// Kernel A: per-batch-row keep-mask scan.
//   keep[t] = !non_visible[t] && !padding[t]
//   src_of[b][j] = t of the j-th kept frame (stable order), -1 for j >= len
//   also writes visible_padding_mask (0/1 as float) and visible_lens (float)
// One 256-thread block per batch row; each thread owns 8 consecutive frames,
// loaded as a single 8-byte word per mask (branch-free keep computation).
// ---------------------------------------------------------------------------
__global__ __launch_bounds__(256) void patchout_scan_kernel(
    const unsigned char* __restrict__ non_visible,   // [B,T] numpy bool (1B)
    const unsigned char* __restrict__ padding,       // [B,T] numpy bool (1B)
    int*   __restrict__ src_of,                      // [B,T] scratch
    float* __restrict__ out_mask,                    // [B,T] -> d_out tail
    float* __restrict__ out_lens)                    // [B]   -> d_out tail
{
    const int b   = blockIdx.x;
    const int tid = threadIdx.x;
    __shared__ int cnt[256];

    const int base = tid * 8;   // 8-byte aligned (rows are 2048 B)
    const uint2 nvv = *(const uint2*)(non_visible + b * PT + base);
    const uint2 pdv = *(const uint2*)(padding     + b * PT + base);
    // byte e is "keep" iff both mask bytes are zero <=> OR'd byte is zero
    const uint64_t m = (((uint64_t)nvv.y << 32) | nvv.x) |
                       (((uint64_t)pdv.y << 32) | pdv.x);

    int keep[8];
    int c = 0;
#pragma unroll
    for (int e = 0; e < 8; ++e) {
        const int k = (((m >> (8 * e)) & 0xffull) == 0) ? 1 : 0;
        keep[e] = k;
        c += k;
    }
    cnt[tid] = c;
    __syncthreads();

    // Hillis-Steele inclusive scan over the 256 per-thread counts
    for (int off = 1; off < 256; off <<= 1) {
        const int v = (tid >= off) ? cnt[tid - off] : 0;
        __syncthreads();
        cnt[tid] += v;
        __syncthreads();
    }
    const int total = cnt[255];        // visible length for this row
    int run = cnt[tid] - c;            // exclusive prefix for this chunk

#pragma unroll
    for (int e = 0; e < 8; ++e) {
        if (keep[e]) {
            src_of[b * PT + run] = base + e;   // slots [0, total) — disjoint
            ++run;
        }
    }
    // Tail fill (slots [total, T)) + padding mask + length
    for (int j = tid; j < PT; j += 256) {
        const int padded = (j >= total);
        if (padded) src_of[b * PT + j] = -1;
        out_mask[b * PT + j] = padded ? 1.0f : 0.0f;
    }
    if (tid == 0) out_lens[b] = (float)total;
}

// ---------------------------------------------------------------------------
// Kernel B: row gather + PE add. One wave32 per output row (row = 768 floats
// = 3 KB). The wave stages the source row into its private LDS slice with
// 6 x global_load_async_to_lds_b128 (32 lanes x 16B each), computes the 24
// sinusoidal PE values per lane while the copy is in flight (chunks c and
// c+3 share the same angle: sin/cos halves), then waits ASYNCcnt and streams
// 128-bit stores. Pure HBM-bound: ~200 MB total -> ~8.7 us at 23.3 TB/s.
// ---------------------------------------------------------------------------
__global__ __launch_bounds__(256) void patchout_gather_kernel(
    const float* __restrict__ in,        // [B,T,D]
    const int*   __restrict__ src_of,    // [B,T]
    float*       __restrict__ out)       // [B,T,D]
{
    constexpr int WAVES = 8;                         // 256 threads / wave32
    __shared__ __align__(16) float buf[WAVES * PD];  // 24 KB (of 320 KB/WGP)

    const int wave = threadIdx.x >> 5;
    const int lane = threadIdx.x & 31;
    const int row  = blockIdx.x * WAVES + wave;    // 0 .. B*T-1 (wave-uniform)
    const int b    = row >> 11;                    // / T
    const int s    = src_of[row];                  // wave-uniform source frame

    float4* outv = (float4*)(out + (size_t)row * PD);

    if (s < 0) {  // zero-padded tail row (uniform branch: EXEC stays full)
        const float4 z = make_float4(0.f, 0.f, 0.f, 0.f);
#pragma unroll
        for (int c = 0; c < 6; ++c) outv[c * 32 + lane] = z;
        return;
    }

    const float* src  = in + ((size_t)b * PT + (size_t)s) * PD;
    float*       wbuf = buf + wave * PD;

    // 1) Kick off the async copy of the whole row into LDS.
#pragma unroll
    for (int c = 0; c < 6; ++c) {
        const int elt = c * 128 + lane * 4;
        async_load_b128_to_lds((uint32_t)(uintptr_t)(wbuf + elt), src + elt);
    }

    // 2) Overlap: sinusoidal PE.  pe[d] = sin(pos * 10000^(-d/384)) for d<384,
    //    cos(...) with i=d-384 otherwise.  Chunk c (sin) and c+3 (cos) share i.
    const float pos     = (float)s;
    const float c_exp2  = -13.287712379549449f / 384.0f;   // -log2(1e4)/384
    const float inv2pi  = 0.15915494309189535f;
    const float twopi   = 6.283185307179586f;
    float pe[6][4];
#pragma unroll
    for (int c = 0; c < 3; ++c) {
#pragma unroll
        for (int k = 0; k < 4; ++k) {
            const int   i   = c * 128 + lane * 4 + k;      // i in [0,384)
            const float inv = __builtin_amdgcn_exp2f(c_exp2 * (float)i);
            float f = pos * inv * inv2pi;                  // turns
            f -= rintf(f);                                 // [-0.5, 0.5]
            const float r = f * twopi;                     // [-pi, pi]
            pe[c][k]     = __sinf(r);
            pe[c + 3][k] = __cosf(r);
        }
    }

    // 3) Wait for the async copy, then add PE and stream out.
    wait_async_cnt0();
#pragma unroll
    for (int c = 0; c < 6; ++c) {
        const int elt = c * 128 + lane * 4;
        float4 v = *(const float4*)(wbuf + elt);           // ds_load_b128
        v.x += pe[c][0];
        v.y += pe[c][1];
        v.z += pe[c][2];
        v.w += pe[c][3];
        outv[c * 32 + lane] = v;                           // global_store_b128
    }
}

// ---------------------------------------------------------------------------
// d_out layout (concatenated, fp32): visible [B,T,D] | mask [B,T] | lens [B]
// d_ws: src_of int[B*T] (128 KB)
// ---------------------------------------------------------------------------
extern "C" void kernel_launch(void* const* d_in, const int* in_sizes, int n_in,
                              void* d_out, int out_size, void* d_ws, size_t ws_size,
                              hipStream_t stream) {
    (void)in_sizes; (void)n_in; (void)out_size; (void)ws_size;

    const float*         wav    = (const float*)d_in[0];
    const unsigned char* nonvis = (const unsigned char*)d_in[1]; // numpy bool
    const unsigned char* padm   = (const unsigned char*)d_in[2]; // numpy bool

    float* out      = (float*)d_out;
    float* out_mask = out + (size_t)PB * PT * PD;
    float* out_lens = out_mask + (size_t)PB * PT;
    int*   src_of   = (int*)d_ws;

    patchout_scan_kernel<<<PB, 256, 0, stream>>>(nonvis, padm, src_of,
                                                 out_mask, out_lens);
    patchout_gather_kernel<<<(PB * PT) / 8, 256, 0, stream>>>(wav, src_of, out);
}